// End2End_10316511445013
// MI455X (gfx1250) — compile-verified
//
#include <hip/hip_runtime.h>
#include <hip/hip_bf16.h>

// Problem constants (from reference)
#define BB 4
#define LL 512
#define VV 32128
#define AVV 32000
#define DD 768

// ---------------------------------------------------------------------------
// key = logit - ln(-ln(u)), computed with raw v_log_f32 (base-2 log, no
// denormal fixup -- inputs are provably normal: u in [1e-6, 1-1e-6]).
//   -ln(u)        = -ln2 * log2(u)
//   ln(-ln(u))    =  ln2 * log2(-ln2 * log2(u))
// ---------------------------------------------------------------------------
__device__ __forceinline__ float gumbel_key(float a, float u)
{
    const float LN2 = 0.69314718055994530942f;
    const float t = -LN2 * __builtin_amdgcn_logf(u);           // = -ln(u) > 0
    return __builtin_fmaf(-LN2, __builtin_amdgcn_logf(t), a);  // a - ln(-ln(u))
}

// ---------------------------------------------------------------------------
// Kernel 1: per-row argmax of (logits - log(-log(u))).
// TAU == 1 and softmax is monotone, so argmax of y_soft == argmax of this key.
// 2048 rows, each 32128 f32 x 2 tensors -> 526 MB streamed: bandwidth bound.
// ---------------------------------------------------------------------------
__global__ __launch_bounds__(256) void argmax_rows(
    const float* __restrict__ logits,
    const float* __restrict__ gumbel,
    int* __restrict__ hard_idx)
{
    const int row = blockIdx.x;                 // 0 .. B*L-1
    const int tid = threadIdx.x;
    const size_t base = (size_t)row * VV;

    const float4* __restrict__ lg4 = (const float4*)(logits + base);
    const float4* __restrict__ gu4 = (const float4*)(gumbel + base);
    // n4 = VV/4 = 8032 float4 elements per row.
    // Main loop: 15 uniform iterations x (2 float4 per thread) covers
    // per-thread indices tid + 256*m for m = 0..29, i.e. [0, 7680).
    // Tail covers [7680, 8032) = one full-block step + one tid<96 step.

    float bestv = -__builtin_inff();
    int   besti = 0x7fffffff;

    // Process one float4 pair at flat index i (indices ascend across calls,
    // so strictly-greater keeps the first/lowest index within this thread).
    auto step = [&](int i, float4 a, float4 u) {
        float k0 = gumbel_key(a.x, u.x);
        float k1 = gumbel_key(a.y, u.y);
        float k2 = gumbel_key(a.z, u.z);
        float k3 = gumbel_key(a.w, u.w);
        const int ib = i * 4;
        if (k0 > bestv) { bestv = k0; besti = ib + 0; }
        if (k1 > bestv) { bestv = k1; besti = ib + 1; }
        if (k2 > bestv) { bestv = k2; besti = ib + 2; }
        if (k3 > bestv) { bestv = k3; besti = ib + 3; }
    };

    // ---- uniform main loop: no guards, 4 independent b128 loads in flight
    for (int k = 0; k < 15; ++k) {
        const int i = tid + k * 512;
        const int j = i + 256;

        // gfx1250 global_prefetch_b8: stream-ahead into GL2 (~64 KB ahead).
        __builtin_prefetch(lg4 + i + 4096, 0, 0);
        __builtin_prefetch(gu4 + i + 4096, 0, 0);

        const float4 a0 = lg4[i];
        const float4 u0 = gu4[i];
        const float4 a1 = lg4[j];
        const float4 u1 = gu4[j];

        step(i, a0, u0);
        step(j, a1, u1);
    }

    // ---- tail: [7680, 8032)
    {
        const int i = 7680 + tid;                 // 7680..7935, all in-bounds
        step(i, lg4[i], gu4[i]);
    }
    if (tid < 96) {
        const int i = 7936 + tid;                 // 7936..8031
        step(i, lg4[i], gu4[i]);
    }

    __shared__ float sv[256];
    __shared__ int   si[256];
    sv[tid] = bestv;
    si[tid] = besti;
    __syncthreads();

    #pragma unroll
    for (int s = 128; s > 0; s >>= 1) {
        if (tid < s) {
            const float v2 = sv[tid + s];
            const int   i2 = si[tid + s];
            // first-index tie-break, matching jnp.argmax
            if (v2 > sv[tid] || (v2 == sv[tid] && i2 < si[tid])) {
                sv[tid] = v2;
                si[tid] = i2;
            }
        }
        __syncthreads();
    }
    if (tid == 0) hard_idx[row] = si[0];
}

// ---------------------------------------------------------------------------
// Kernel 2: per-batch psg preparation.
//   len[b]        = sum_l rwrt[b,l]
//   psg[b,p]      = (p==0) ? 1 : psg_input[b,p-1]            (roll by 1)
//   extr[b,p]     = (1 - rwrt[b, L-1-p]) * psg[b,p]          (fliplr mask)
//   trunc[b,l]    = extr[b, (l - len[b]) mod L]              (take_along_axis)
//   firstnz[b]    = min { l : trunc[b,l] != 0 }  (flag[l] == l >= firstnz)
// ---------------------------------------------------------------------------
__global__ __launch_bounds__(256) void prep_psg(
    const int* __restrict__ rwrt,
    const int* __restrict__ psg_in,
    int* __restrict__ trunc_ids,
    int* __restrict__ firstnz)
{
    const int b   = blockIdx.x;
    const int tid = threadIdx.x;

    __shared__ int ssum[256];
    __shared__ int s_len;
    __shared__ int s_first;

    int acc = 0;
    for (int l = tid; l < LL; l += 256) acc += rwrt[b * LL + l];
    ssum[tid] = acc;
    __syncthreads();
    #pragma unroll
    for (int s = 128; s > 0; s >>= 1) {
        if (tid < s) ssum[tid] += ssum[tid + s];
        __syncthreads();
    }
    if (tid == 0) { s_len = ssum[0]; s_first = LL; }
    __syncthreads();

    const int len = s_len;
    for (int l = tid; l < LL; l += 256) {
        const int p  = (l - len) & (LL - 1);               // L is a power of 2
        const int fm = 1 - rwrt[b * LL + (LL - 1 - p)];
        const int ps = (p == 0) ? 1 : psg_in[b * LL + p - 1];
        const int t  = fm * ps;
        trunc_ids[b * LL + l] = t;
        if (t != 0) atomicMin(&s_first, l);
    }
    __syncthreads();
    if (tid == 0) firstnz[b] = s_first;
}

// ---------------------------------------------------------------------------
// Kernel 3: output assembly. One block per (b,l) row, one float4 of D per
// thread (D = 768 = 192 float4).
//   out = mask * (idx1<AV ? W[idx1] : 0)  +  (l>=firstnz ? W[trunc] : 0)
// W (98 MB) lives in the 192 MB L2 after first touch; total gather ~12 MB.
// ---------------------------------------------------------------------------
__global__ __launch_bounds__(192) void gather_out(
    const float* __restrict__ W,
    const int*   __restrict__ rwrt,
    const int*   __restrict__ hard_idx,
    const int*   __restrict__ trunc_ids,
    const int*   __restrict__ firstnz,
    float* __restrict__ out)
{
    const int row = blockIdx.x;          // b*L + l
    const int b   = row >> 9;            // / 512
    const int l   = row & (LL - 1);
    const int t   = threadIdx.x;         // 0..191

    const float4* __restrict__ W4 = (const float4*)W;
    float4 acc = make_float4(0.f, 0.f, 0.f, 0.f);

    const int i1 = hard_idx[row];
    if (rwrt[row] != 0 && i1 < AVV) {
        acc = W4[(size_t)i1 * (DD / 4) + t];
    }
    if (l >= firstnz[b]) {
        const int i2 = trunc_ids[row];   // in [0, AV); flag may select W[0] row
        const float4 e = W4[(size_t)i2 * (DD / 4) + t];
        acc.x += e.x; acc.y += e.y; acc.z += e.z; acc.w += e.w;
    }
    ((float4*)out)[(size_t)row * (DD / 4) + t] = acc;
}

// ---------------------------------------------------------------------------
extern "C" void kernel_launch(void* const* d_in, const int* in_sizes, int n_in,
                              void* d_out, int out_size, void* d_ws, size_t ws_size,
                              hipStream_t stream)
{
    (void)in_sizes; (void)n_in; (void)out_size; (void)ws_size;

    const float* logits = (const float*)d_in[0];   // (B,L,V) f32
    const float* gumbel = (const float*)d_in[1];   // (B,L,V) f32
    const float* W      = (const float*)d_in[2];   // (AV,D)  f32
    const int*   rwrt   = (const int*)  d_in[3];   // (B,L)   i32
    const int*   psg    = (const int*)  d_in[4];   // (B,L)   i32
    float*       out    = (float*)      d_out;     // (B,L,D) f32

    int* ws        = (int*)d_ws;
    int* hard_idx  = ws;                 // B*L ints
    int* trunc_ids = ws + BB * LL;       // B*L ints
    int* firstnz   = ws + 2 * BB * LL;   // B ints

    argmax_rows<<<BB * LL, 256, 0, stream>>>(logits, gumbel, hard_idx);
    prep_psg  <<<BB,       256, 0, stream>>>(rwrt, psg, trunc_ids, firstnz);
    gather_out<<<BB * LL,  192, 0, stream>>>(W, rwrt, hard_idx, trunc_ids, firstnz, out);
}